// MNN_GNN_47218870452456
// MI455X (gfx1250) — compile-verified
//
#include <hip/hip_runtime.h>

#define N_NODES 100000
#define N_EDGES 625000
#define DIM     128
#define HID     64
#define NCLS    6

typedef __attribute__((ext_vector_type(2))) float v2f;
typedef __attribute__((ext_vector_type(8))) float v8f;

__device__ __forceinline__ float leaky01(float v) { return v >= 0.0f ? v : 0.1f * v; }

// ---------------- degree ----------------
__global__ void k_deg_init(float* __restrict__ deg) {
    int i = blockIdx.x * 256 + threadIdx.x;
    if (i < N_NODES) deg[i] = 1.0f;   // +1 self loop
}

__global__ void k_deg_accum(const int* __restrict__ ei, float* __restrict__ deg) {
    int e = blockIdx.x * 256 + threadIdx.x;
    if (e < N_EDGES) {
        int d = ei[N_EDGES + e];      // dst row
        atomicAdd(&deg[d], 1.0f);
    }
}

// ---------------- GEMM1: H[N,128] = X[N,128] @ W[128,128] (f32 WMMA) ----------------
// one wave per 16x16 output tile; 6250 row tiles x 8 col tiles = 50000 waves
__global__ void k_gemm_xW(const float* __restrict__ X, const float* __restrict__ W,
                          float* __restrict__ H) {
    int wave = (blockIdx.x * blockDim.x + threadIdx.x) >> 5;
    int lane = threadIdx.x & 31;
    int rowTile = wave >> 3;          // 0..6249
    int colTile = wave & 7;           // 0..7
    int row0 = rowTile * 16, col0 = colTile * 16;
    int nl = lane & 15;               // M for A rows, N for B/C cols
    int hi = lane >> 4;               // lane-half selector

    v8f c = {};
    const float* arow = X + (long)(row0 + nl) * DIM + 2 * hi;      // A: row M, K=2*hi..2*hi+1
    const float* bcol = W + (long)(2 * hi) * DIM + col0 + nl;      // B: K=2*hi rows, col N
    for (int kk = 0; kk < DIM; kk += 4) {
        v2f a = *(const v2f*)(arow + kk);
        v2f b;
        b.x = bcol[(long)kk * DIM];
        b.y = bcol[(long)(kk + 1) * DIM];
        c = __builtin_amdgcn_wmma_f32_16x16x4_f32(false, a, false, b,
                                                  (short)0, c, false, false);
    }
    float* out = H + col0 + nl;
#pragma unroll
    for (int i = 0; i < 8; ++i)
        out[(long)(row0 + i + 8 * hi) * DIM] = c[i];
}

// ---------------- agg init: self-loop term + bias ----------------
__global__ void k_agg_init(const float4* __restrict__ h4, const float* __restrict__ deg,
                           const float* __restrict__ b1, float4* __restrict__ agg4) {
    int i = blockIdx.x * 256 + threadIdx.x;           // float4 index over N*128
    if (i >= N_NODES * (DIM / 4)) return;
    int node = i >> 5;                                // i / 32
    int d4   = i & 31;
    float inv = 1.0f / deg[node];                     // dinv*dinv = 1/deg
    float4 hv = h4[i];
    float4 bv = ((const float4*)b1)[d4];
    float4 o;
    o.x = hv.x * inv + bv.x;
    o.y = hv.y * inv + bv.y;
    o.z = hv.z * inv + bv.z;
    o.w = hv.w * inv + bv.w;
    agg4[i] = o;
}

// ---------------- edge scatter: agg[dst] += h[src] * rsqrt(deg_s)*rsqrt(deg_d) ----------------
// 32 lanes per edge, float4 per lane
__global__ void k_edge_scatter(const int* __restrict__ ei, const float* __restrict__ deg,
                               const float* __restrict__ h, float* __restrict__ agg) {
    int t = blockIdx.x * 256 + threadIdx.x;
    int e = t >> 5;
    if (e >= N_EDGES) return;
    int lane = t & 31;
    int s = ei[e];
    int d = ei[N_EDGES + e];
    float w = rsqrtf(deg[s]) * rsqrtf(deg[d]);
    float4 v = *((const float4*)(h + (long)s * DIM) + lane);  // L2-resident gather
    float* ap = agg + (long)d * DIM + lane * 4;
    atomicAdd(ap + 0, v.x * w);
    atomicAdd(ap + 1, v.y * w);
    atomicAdd(ap + 2, v.z * w);
    atomicAdd(ap + 3, v.w * w);
}

// ---------------- elementwise chain: z = x + 0.01*leaky(BN(leaky(leaky(agg)))) ----------------
__global__ void k_post(const float* __restrict__ x, const float* __restrict__ agg,
                       const float* __restrict__ gamma, const float* __restrict__ beta,
                       const float* __restrict__ mean, const float* __restrict__ var,
                       float* __restrict__ z) {
    int i = blockIdx.x * 256 + threadIdx.x;
    if (i >= N_NODES * DIM) return;
    int dcol = i & (DIM - 1);
    float x1 = leaky01(agg[i]);                                   // gcn output activation
    float zz = leaky01(x1);                                       // second leaky
    zz = (zz - mean[dcol]) * rsqrtf(var[dcol] + 1e-5f) * gamma[dcol] + beta[dcol];
    zz = leaky01(zz);
    z[i] = x[i] + 0.01f * zz;                                     // residual
}

// ---------------- GEMM2: T[N,64] = leaky(Z[N,128] @ lin1_W^T + lin1_b) ----------------
// B[k][n] = lin1_W[n*128 + k]  -> per-lane contiguous float2 in k
__global__ void k_gemm_lin1(const float* __restrict__ Z, const float* __restrict__ Wt,
                            const float* __restrict__ bias, float* __restrict__ T) {
    int wave = (blockIdx.x * blockDim.x + threadIdx.x) >> 5;
    int lane = threadIdx.x & 31;
    int rowTile = wave >> 2;          // 0..6249
    int colTile = wave & 3;           // 0..3
    int row0 = rowTile * 16, col0 = colTile * 16;
    int nl = lane & 15;
    int hi = lane >> 4;

    v8f c = {};
    const float* arow = Z  + (long)(row0 + nl) * DIM + 2 * hi;
    const float* brow = Wt + (long)(col0 + nl) * DIM + 2 * hi;    // transposed B
    for (int kk = 0; kk < DIM; kk += 4) {
        v2f a = *(const v2f*)(arow + kk);
        v2f b = *(const v2f*)(brow + kk);
        c = __builtin_amdgcn_wmma_f32_16x16x4_f32(false, a, false, b,
                                                  (short)0, c, false, false);
    }
    float bn = bias[col0 + nl];
    float* out = T + col0 + nl;
#pragma unroll
    for (int i = 0; i < 8; ++i)
        out[(long)(row0 + i + 8 * hi) * HID] = leaky01(c[i] + bn);
}

// ---------------- head: logits -> softmax -> cumprod(1-p) ----------------
__global__ void k_head(const float* __restrict__ T, const float* __restrict__ W3,
                       const float* __restrict__ b3, float* __restrict__ out) {
    int n = blockIdx.x * 256 + threadIdx.x;
    if (n >= N_NODES) return;
    float t[HID];
    const float4* tp = (const float4*)(T + (long)n * HID);
#pragma unroll
    for (int i = 0; i < HID / 4; ++i) {
        float4 v = tp[i];
        t[4 * i + 0] = v.x; t[4 * i + 1] = v.y; t[4 * i + 2] = v.z; t[4 * i + 3] = v.w;
    }
    float logits[NCLS];
#pragma unroll
    for (int j = 0; j < NCLS; ++j) {
        float acc = b3[j];
        const float* w = W3 + j * HID;
#pragma unroll
        for (int k = 0; k < HID; ++k) acc += t[k] * w[k];
        logits[j] = acc;
    }
    float mx = logits[0];
#pragma unroll
    for (int j = 1; j < NCLS; ++j) mx = fmaxf(mx, logits[j]);
    float p[NCLS]; float sum = 0.0f;
#pragma unroll
    for (int j = 0; j < NCLS; ++j) { p[j] = __expf(logits[j] - mx); sum += p[j]; }
    float inv = 1.0f / sum;
    float* pred = out + (long)n * NCLS;
    float* Sout = out + (long)N_NODES * NCLS + (long)n * NCLS;
    float S = 1.0f;
#pragma unroll
    for (int j = 0; j < NCLS; ++j) {
        float pj = p[j] * inv;
        pred[j] = pj;
        S *= (1.0f - pj);
        Sout[j] = S;
    }
}

extern "C" void kernel_launch(void* const* d_in, const int* in_sizes, int n_in,
                              void* d_out, int out_size, void* d_ws, size_t ws_size,
                              hipStream_t stream) {
    const float* x      = (const float*)d_in[0];
    const float* W1     = (const float*)d_in[1];
    const float* b1     = (const float*)d_in[2];
    const float* bn_g   = (const float*)d_in[3];
    const float* bn_b   = (const float*)d_in[4];
    const float* bn_m   = (const float*)d_in[5];
    const float* bn_v   = (const float*)d_in[6];
    const float* lin1_W = (const float*)d_in[7];
    const float* lin1_b = (const float*)d_in[8];
    const float* lin3_W = (const float*)d_in[9];
    const float* lin3_b = (const float*)d_in[10];
    const int*   ei     = (const int*)d_in[11];
    float* out = (float*)d_out;

    // workspace layout (floats): deg[N] | h[N*128] | agg[N*128]
    float* ws  = (float*)d_ws;
    float* deg = ws;
    float* h   = ws + N_NODES;
    float* agg = h + (long)N_NODES * DIM;
    float* z   = h;    // reuse: h dead after edge scatter
    float* t   = agg;  // reuse: agg dead after k_post

    k_deg_init   <<<(N_NODES + 255) / 256, 256, 0, stream>>>(deg);
    k_deg_accum  <<<(N_EDGES + 255) / 256, 256, 0, stream>>>(ei, deg);
    k_gemm_xW    <<<(N_NODES / 16) * 8 / 8, 256, 0, stream>>>(x, W1, h);          // 6250 blocks
    k_agg_init   <<<(N_NODES * (DIM / 4) + 255) / 256, 256, 0, stream>>>(
                     (const float4*)h, deg, b1, (float4*)agg);
    k_edge_scatter<<<(N_EDGES * 32 + 255) / 256, 256, 0, stream>>>(ei, deg, h, agg);
    k_post       <<<(N_NODES * DIM + 255) / 256, 256, 0, stream>>>(
                     x, agg, bn_g, bn_b, bn_m, bn_v, z);
    k_gemm_lin1  <<<(N_NODES / 16) * 4 / 8, 256, 0, stream>>>(z, lin1_W, lin1_b, t); // 3125 blocks
    k_head       <<<(N_NODES + 255) / 256, 256, 0, stream>>>(t, lin3_W, lin3_b, out);
}